// Mute_78125455115031
// MI455X (gfx1250) — compile-verified
//
#include <hip/hip_runtime.h>

typedef __attribute__((ext_vector_type(2))) float v2f;
typedef __attribute__((ext_vector_type(8))) float v8f;

#define HH 96
#define WW 96
#define NPIX (HH * WW)      // 9216 pixels per channel
#define CCH 256             // channels (= K = output channels)
#define NCOL NPIX           // GEMM N dimension
#define KDIM CCH            // GEMM K dimension
#define CW (CCH * WW)       // 24576 = width of the [B*H, C*W] reshape

// ---------------------------------------------------------------------------
// Kernel 1: per-channel "frequency filter" reduced to a 16-coefficient DFT.
// q = floor(255*(|x|-min)/ptp); keep only frequencies (ky,kx) in {-2,-1,0,1}^2
// (the fftshift-centered 4x4 block); FQ = |inverse DFT of those 16 coeffs|.
// One block per channel; q lives in LDS (36 KB).
// ---------------------------------------------------------------------------
__global__ __launch_bounds__(256) void freq_q_kernel(const float* __restrict__ x,
                                                     float* __restrict__ FQ) {
  const int c = blockIdx.x;
  const float* xc = x + (size_t)c * NPIX;
  __shared__ float q[NPIX];
  __shared__ float red[256];
  __shared__ float Gre[4 * WW], Gim[4 * WW];
  __shared__ float Fre[16], Fim[16];
  const int t = threadIdx.x;
  const float W0 = 6.2831853071795864769f / 96.0f;  // 2*pi/96

  // per-channel min / max of |x|
  float mn = 3.4e38f, mx = -3.4e38f;
  for (int i = t; i < NPIX; i += 256) {
    float a = fabsf(xc[i]);
    mn = fminf(mn, a);
    mx = fmaxf(mx, a);
  }
  red[t] = mn; __syncthreads();
  for (int s = 128; s > 0; s >>= 1) { if (t < s) red[t] = fminf(red[t], red[t + s]); __syncthreads(); }
  mn = red[0]; __syncthreads();
  red[t] = mx; __syncthreads();
  for (int s = 128; s > 0; s >>= 1) { if (t < s) red[t] = fmaxf(red[t], red[t + s]); __syncthreads(); }
  mx = red[0]; __syncthreads();
  const float inv = 255.0f / (mx - mn);

  for (int i = t; i < NPIX; i += 256)
    q[i] = floorf((fabsf(xc[i]) - mn) * inv);
  __syncthreads();

  // G[ky][x] = sum_y q[y][x] * e^{-i*W0*(ky-2)*y}
  for (int p = t; p < 4 * WW; p += 256) {
    const int ky = p / WW;
    const int xx = p - ky * WW;
    const float kf = (float)(ky - 2);
    float sre = 0.f, sim = 0.f;
    for (int y = 0; y < HH; ++y) {
      float sv, cv;
      __sincosf(-W0 * kf * (float)y, &sv, &cv);
      const float v = q[y * WW + xx];
      sre = fmaf(v, cv, sre);
      sim = fmaf(v, sv, sim);
    }
    Gre[p] = sre; Gim[p] = sim;
  }
  __syncthreads();

  // F[ky][kx] = sum_x G[ky][x] * e^{-i*W0*(kx-2)*x}
  if (t < 16) {
    const int ky = t >> 2, kx = t & 3;
    const float kf = (float)(kx - 2);
    float sre = 0.f, sim = 0.f;
    for (int xx = 0; xx < WW; ++xx) {
      float sv, cv;
      __sincosf(-W0 * kf * (float)xx, &sv, &cv);
      const float gr = Gre[ky * WW + xx], gi = Gim[ky * WW + xx];
      sre += gr * cv - gi * sv;
      sim += gr * sv + gi * cv;
    }
    Fre[t] = sre; Fim[t] = sim;
  }
  __syncthreads();

  // FQ[y][x] = |(1/9216) * sum_{16} F * e^{+i*W0*((ky-2)*y + (kx-2)*x)}|
  const float normf = 1.0f / (float)NPIX;
  for (int i = t; i < NPIX; i += 256) {
    const int y = i / WW, xx = i - y * WW;
    float sre = 0.f, sim = 0.f;
#pragma unroll
    for (int k = 0; k < 16; ++k) {
      const int ky = (k >> 2) - 2, kx = (k & 3) - 2;
      float sv, cv;
      __sincosf(W0 * (float)(ky * y + kx * xx), &sv, &cv);
      sre += Fre[k] * cv - Fim[k] * sv;
      sim += Fre[k] * sv + Fim[k] * cv;
    }
    sre *= normf; sim *= normf;
    FQ[(size_t)c * NPIX + i] = sqrtf(sre * sre + sim * sim);
  }
}

// ---------------------------------------------------------------------------
// Kernel 2: Y[256,9216] = Wm[256,256] @ X[256,9216] + bias, via
// V_WMMA_F32_16X16X4_F32 (exact fp32). One 16x16 output tile per wave,
// 8 waves per block. If APPLY_MASK: Y *= (1 + colmax[flat % 24576]), where the
// mod is computed cheaply: m*9216 mod 24576 == 3072*((3m)&7)  (9216=3*3072,
// 24576=8*3072), plus one conditional subtract — no 64-bit division chain.
// A layout (16x4 f32): lane<16 -> m=lane, k={0,1}; lane>=16 -> m=lane-16, k={2,3}.
// B layout (4x16 f32): lane<16 -> n=lane, k={0,1}; lane>=16 -> n=lane-16, k={2,3}.
// C/D: VGPR v, lane l -> M = v + 8*(l>>4), N = l&15.
// ---------------------------------------------------------------------------
template <bool APPLY_MASK>
__global__ __launch_bounds__(256) void gemm16_wmma(const float* __restrict__ Wm,
                                                   const float* __restrict__ X,
                                                   const float* __restrict__ bias,
                                                   float* __restrict__ Y,
                                                   const float* __restrict__ colmax) {
  const int lane = threadIdx.x & 31;
  const int wave = threadIdx.x >> 5;
  const int half = lane >> 4;
  const int l15 = lane & 15;
  const int m0 = blockIdx.y * 16;
  const int n0 = (blockIdx.x * 8 + wave) * 16;

  const float* arow = Wm + (size_t)(m0 + l15) * KDIM + 2 * half;
  const float* bptr = X + (size_t)(2 * half) * NCOL + n0 + l15;

  v8f acc = {};
#pragma unroll 4
  for (int k = 0; k < KDIM; k += 4) {
    v2f a, b;
    a.x = arow[k];
    a.y = arow[k + 1];
    b.x = bptr[(size_t)k * NCOL];
    b.y = bptr[(size_t)k * NCOL + NCOL];
    acc = __builtin_amdgcn_wmma_f32_16x16x4_f32(false, a, false, b,
                                                (short)0, acc, false, false);
  }

  const int n = n0 + l15;
#pragma unroll
  for (int v = 0; v < 8; ++v) {
    const int m = m0 + v + 8 * half;
    float val = acc[v] + bias[m];
    if (APPLY_MASK) {
      // j = (m*9216 + n) mod 24576, via 3072*((3m)&7) + n and one cond-sub
      int j = 3072 * ((3 * m) & 7) + n;
      j = (j >= CW) ? (j - CW) : j;
      val *= (1.0f + colmax[j]);
    }
    Y[(size_t)m * NCOL + n] = val;
  }
}

// ---------------------------------------------------------------------------
// Kernel 3: Fuse_Q_mask[j] = max over r=0..95 of Qflat[r*24576 + j]
// (the reshape [1,256,9216] -> [96,24576] followed by max over axis 0).
// ---------------------------------------------------------------------------
__global__ __launch_bounds__(256) void colmax_kernel(const float* __restrict__ Q,
                                                     float* __restrict__ cm) {
  const int j = blockIdx.x * 256 + threadIdx.x;  // grid sized exactly: 96*256 = 24576
  float m = -3.4e38f;
  for (int r = 0; r < HH; ++r)
    m = fmaxf(m, Q[(size_t)r * CW + j]);
  cm[j] = m;
}

extern "C" void kernel_launch(void* const* d_in, const int* in_sizes, int n_in,
                              void* d_out, int out_size, void* d_ws, size_t ws_size,
                              hipStream_t stream) {
  const float* fuse = (const float*)d_in[0];
  const float* Wq   = (const float*)d_in[1];
  const float* bq   = (const float*)d_in[2];
  // d_in[3] = Wk, d_in[4] = bk: provably dead — softmax over the singleton
  // batch axis makes mask_KQ identically 1, so the Key path never matters.
  const float* Wv   = (const float*)d_in[5];
  const float* bv   = (const float*)d_in[6];

  float* FQ = (float*)d_ws;                        // [256, 9216]
  float* Qb = FQ + (size_t)CCH * NPIX;             // [256, 9216]
  float* cm = Qb + (size_t)CCH * NPIX;             // [24576]

  // 1) high-pass (actually low-keep) frequency filter of batch element 0
  freq_q_kernel<<<CCH, 256, 0, stream>>>(fuse, FQ);
  // 2) fuse_Query = Wq @ FQ + bq  (WMMA f32)
  gemm16_wmma<false><<<dim3(72, 16), 256, 0, stream>>>(Wq, FQ, bq, Qb, nullptr);
  // 3) Fuse_Q_mask = column max of [96, 24576] reshape
  colmax_kernel<<<96, 256, 0, stream>>>(Qb, cm);
  // 4) out = (Wv @ fuse + bv) * (1 + broadcast mask)  (WMMA f32, fused epilogue)
  gemm16_wmma<true><<<dim3(72, 16), 256, 0, stream>>>(Wv, fuse, bv, (float*)d_out, cm);
}